// OT_GNN_layer_10977936409019
// MI455X (gfx1250) — compile-verified
//
#include <hip/hip_runtime.h>

// ---------------- problem constants (from reference) ----------------
#define N_NODES   100000
#define DEG       16
#define M_PATCH   17          // center + 16 neighbors
#define N_FEAT    3
#define N_TPL     10
#define TPL_N     5
#define N_CLASSES 3
#define ALPHA     0.5f
#define REG_E     0.1f
#define SINK_ITERS 10
#define MD_ITERS   3
#define EPS_S      1e-16f

// derived
#define INV_M   (1.0f / 17.0f)
#define INV_N   (1.0f / 5.0f)
#define CC1_0   (16.0f / 17.0f)   // (C1^2 @ p)[0]
#define CC1_R   (1.0f  / 17.0f)   // (C1^2 @ p)[i>=1]
#define AR      (ALPHA / REG_E)          // 5.0
#define BR_     ((1.0f - ALPHA) / REG_E) // 5.0

typedef __attribute__((ext_vector_type(2))) float v2f;
typedef __attribute__((ext_vector_type(8))) float v8f;

// =====================================================================
// Kernel 1: one thread per (node, template). FGW distance d[node][t].
// Star-graph collapse: the GW tensor has only 2 distinct rows, so
//   K = KM .* w_row,  KM = exp(-(ALPHA/REG)*M)  computed ONCE,
// and each MD iteration only needs 10 fresh exps (w0, wr).
// 128-thread blocks + launch_bounds so the allocator can keep
// KM[17][5] + sinkhorn state (~160 regs) fully in VGPRs (no spill).
// =====================================================================
__global__ __launch_bounds__(128)
void fgw_dist_kernel(const float* __restrict__ x,         // [N,3]
                     const int*   __restrict__ edge_dst,  // [N*DEG]
                     const float* __restrict__ tplC,      // [10,5,5]
                     const float* __restrict__ tplF,      // [10,5,3]
                     float*       __restrict__ dist)      // [N,10]
{
    __shared__ float sC [N_TPL][TPL_N][TPL_N];   // template adjacency
    __shared__ float sF [N_TPL][TPL_N][N_FEAT];  // template features
    __shared__ float sCc[N_TPL][TPL_N];          // cC2[t][n] = (1/5) sum_j C2[n,j]^2
    __shared__ float sFn[N_TPL][TPL_N];          // ||F[t][n]||^2

    for (int i = threadIdx.x; i < N_TPL*TPL_N*TPL_N; i += blockDim.x)
        ((float*)sC)[i] = tplC[i];
    for (int i = threadIdx.x; i < N_TPL*TPL_N*N_FEAT; i += blockDim.x)
        ((float*)sF)[i] = tplF[i];
    __syncthreads();
    if (threadIdx.x < N_TPL*TPL_N) {
        int t = threadIdx.x / TPL_N, nn = threadIdx.x % TPL_N;
        float s2 = 0.f, fn = 0.f;
        #pragma unroll
        for (int j = 0; j < TPL_N; ++j) { float c = sC[t][nn][j]; s2 += c*c; }
        #pragma unroll
        for (int f = 0; f < N_FEAT; ++f) { float v = sF[t][nn][f]; fn += v*v; }
        sCc[t][nn] = s2 * INV_N;
        sFn[t][nn] = fn;
    }
    __syncthreads();

    int gid = blockIdx.x * blockDim.x + threadIdx.x;
    if (gid >= N_NODES * N_TPL) return;
    const int t    = gid % N_TPL;
    const int node = gid / N_TPL;

    // ---- gather patch, build KM = exp(-5*M) (M is rebuilt at the end) ----
    float KM[M_PATCH][TPL_N];
    #pragma unroll
    for (int i = 0; i < M_PATCH; ++i) {
        int src = (i == 0) ? node : edge_dst[node*DEG + (i-1)];
        float px = x[src*3+0], py = x[src*3+1], pz = x[src*3+2];
        float pn = px*px + py*py + pz*pz;
        #pragma unroll
        for (int nn = 0; nn < TPL_N; ++nn) {
            float dot = px*sF[t][nn][0] + py*sF[t][nn][1] + pz*sF[t][nn][2];
            float m   = pn + sFn[t][nn] - 2.0f*dot;
            KM[i][nn] = __expf(-AR * m);
        }
    }

    // ---- mirror descent: T summarized by T0[n]=T[0,:], sR[n]=sum_{i>=1}T[i,:]
    float T0[TPL_N], sR[TPL_N];
    #pragma unroll
    for (int nn = 0; nn < TPL_N; ++nn) { T0[nn] = 1.0f/85.0f; sR[nn] = 16.0f/85.0f; }

    float u[M_PATCH], v[TPL_N], w0[TPL_N], wr[TPL_N];

    #pragma unroll 1
    for (int md = 0; md < MD_ITERS; ++md) {
        // GW tensor has two distinct rows -> two weight vectors
        #pragma unroll
        for (int nn = 0; nn < TPL_N; ++nn) {
            float b0 = 0.f, br = 0.f;
            #pragma unroll
            for (int k = 0; k < TPL_N; ++k) {
                float c2 = sC[t][nn][k];
                b0 += sR[k] * c2;
                br += T0[k] * c2;
            }
            float t0 = CC1_0 + sCc[t][nn] - 2.0f*b0;
            float tr = CC1_R + sCc[t][nn] - 2.0f*br;
            w0[nn] = __expf(-BR_ * t0);
            wr[nn] = __expf(-BR_ * tr);
        }
        // Sinkhorn on K = KM .* w_row (row-class weights folded into v)
        #pragma unroll
        for (int nn = 0; nn < TPL_N; ++nn) v[nn] = 1.0f;
        #pragma unroll 1
        for (int it = 0; it < SINK_ITERS; ++it) {
            float v0[TPL_N], vr[TPL_N];
            #pragma unroll
            for (int nn = 0; nn < TPL_N; ++nn) { v0[nn] = w0[nn]*v[nn]; vr[nn] = wr[nn]*v[nn]; }
            #pragma unroll
            for (int i = 0; i < M_PATCH; ++i) {
                float s = 0.f;
                #pragma unroll
                for (int nn = 0; nn < TPL_N; ++nn)
                    s += KM[i][nn] * ((i == 0) ? v0[nn] : vr[nn]);
                u[i] = INV_M / (s + EPS_S);
            }
            #pragma unroll
            for (int nn = 0; nn < TPL_N; ++nn) {
                float sr = 0.f;
                #pragma unroll
                for (int i = 1; i < M_PATCH; ++i) sr += KM[i][nn] * u[i];
                float cs = KM[0][nn]*w0[nn]*u[0] + wr[nn]*sr;
                v[nn] = INV_N / (cs + EPS_S);
            }
        }
        // summarize T = diag(u) (KM.*w) diag(v) for the next tensor
        #pragma unroll
        for (int nn = 0; nn < TPL_N; ++nn) {
            float sr = 0.f;
            #pragma unroll
            for (int i = 1; i < M_PATCH; ++i) sr += KM[i][nn] * u[i];
            T0[nn] = u[0] * KM[0][nn] * w0[nn] * v[nn];
            sR[nn] = wr[nn] * v[nn] * sr;
        }
    }

    // ---- final cost: <T, ALPHA*M + (1-ALPHA)*tens(T)> ----
    float acc = 0.f;
    #pragma unroll
    for (int nn = 0; nn < TPL_N; ++nn) {
        float b0 = 0.f, br = 0.f;
        #pragma unroll
        for (int k = 0; k < TPL_N; ++k) {
            float c2 = sC[t][nn][k];
            b0 += sR[k] * c2;
            br += T0[k] * c2;
        }
        float t0 = CC1_0 + sCc[t][nn] - 2.0f*b0;
        float tr = CC1_R + sCc[t][nn] - 2.0f*br;
        acc += (1.0f - ALPHA) * (T0[nn]*t0 + sR[nn]*tr);
    }
    // re-gather the patch (L2-resident) to rebuild M exactly
    #pragma unroll
    for (int i = 0; i < M_PATCH; ++i) {
        int src = (i == 0) ? node : edge_dst[node*DEG + (i-1)];
        float px = x[src*3+0], py = x[src*3+1], pz = x[src*3+2];
        float pn = px*px + py*py + pz*pz;
        float rowacc = 0.f;
        #pragma unroll
        for (int nn = 0; nn < TPL_N; ++nn) {
            float dot = px*sF[t][nn][0] + py*sF[t][nn][1] + pz*sF[t][nn][2];
            float m   = pn + sFn[t][nn] - 2.0f*dot;          // M[i][nn]
            float w   = (i == 0) ? w0[nn] : wr[nn];
            rowacc += KM[i][nn] * w * v[nn] * m;             // (T[i,nn]/u[i]) * M
        }
        acc += ALPHA * u[i] * rowacc;
    }
    dist[node*N_TPL + t] = acc;
}

// =====================================================================
// Kernel 2: out = d @ W + b via V_WMMA_F32_16X16X4_F32.
// One wave per 16-node tile; K=10 padded to 3 chunks of 4.
// No divergent loads: A comes from clamped 8B-aligned b64 loads +
// cndmask; B comes from a zero-padded LDS copy of W [12][16].
// =====================================================================
__global__ __launch_bounds__(256)
void proj_wmma_kernel(const float* __restrict__ d,   // [N,10]
                      const float* __restrict__ W,   // [10,3]
                      const float* __restrict__ b,   // [3]
                      float*       __restrict__ out, // [N,3]
                      int ntiles)
{
    __shared__ float sW[12][16];    // zero-padded W: rows k=0..11, cols n=0..15
    for (int i = threadIdx.x; i < 12*16; i += blockDim.x) {
        int r = i >> 4, c = i & 15;
        ((float*)sW)[i] = (r < N_TPL && c < N_CLASSES) ? W[r*N_CLASSES + c] : 0.0f;
    }
    __syncthreads();

    int wave = (blockIdx.x * blockDim.x + threadIdx.x) >> 5;
    int lane = threadIdx.x & 31;
    if (wave >= ntiles) return;          // wave-uniform: EXEC all-ones at WMMA
    int l    = lane & 15;                // row (A) / col (B,C)
    int half = lane >> 4;
    int base = wave * 16;                // first node of tile
    const float* drow = d + (size_t)(base + l) * N_TPL;

    v8f acc = {};
    #pragma unroll
    for (int c0 = 0; c0 < 12; c0 += 4) {
        int k0 = c0 + 2*half;            // A layout: v0 = K 2*half, v1 = K 2*half+1
        int ka = (k0 < 8) ? k0 : 8;      // clamp keeps 8B alignment (even k)
        v2f av = *(const v2f*)(drow + ka);          // unconditional aligned b64
        v2f a;
        a.x = (k0 < N_TPL) ? av.x : 0.0f;           // select, no exec branching
        a.y = (k0 < N_TPL) ? av.y : 0.0f;           // (k0 even: k0<10 <=> k0+1<10)
        v2f bb;
        bb.x = sW[k0    ][l];                       // padded: unconditional
        bb.y = sW[k0 + 1][l];
        acc = __builtin_amdgcn_wmma_f32_16x16x4_f32(
                  /*neg_a=*/false, a, /*neg_b=*/false, bb,
                  /*c_mod=*/(short)0, acc, /*reuse_a=*/false, /*reuse_b=*/false);
    }
    if (l < N_CLASSES) {
        float bias = b[l];
        #pragma unroll
        for (int r = 0; r < 8; ++r) {
            int row = r + 8*half;        // C/D layout: vgpr r = M r / M r+8
            out[(size_t)(base + row)*N_CLASSES + l] = acc[r] + bias;
        }
    }
}

// =====================================================================
extern "C" void kernel_launch(void* const* d_in, const int* in_sizes, int n_in,
                              void* d_out, int out_size, void* d_ws, size_t ws_size,
                              hipStream_t stream)
{
    const float* x    = (const float*)d_in[0];
    const int*   ei   = (const int*)  d_in[1];   // [2, N*DEG]
    const float* tplC = (const float*)d_in[2];
    const float* tplF = (const float*)d_in[3];
    const float* W    = (const float*)d_in[4];
    const float* bias = (const float*)d_in[5];
    float*       out  = (float*)d_out;
    float*       dist = (float*)d_ws;            // [N, 10] scratch (4 MB)

    const int* edge_dst = ei + (size_t)N_NODES * DEG;  // edge_index[1]

    {
        int total = N_NODES * N_TPL;
        int blk = 128;
        fgw_dist_kernel<<<(total + blk - 1) / blk, blk, 0, stream>>>(
            x, edge_dst, tplC, tplF, dist);
    }
    {
        int ntiles = N_NODES / 16;               // 6250, exact
        int waves_per_blk = 256 / 32;
        int grid = (ntiles + waves_per_blk - 1) / waves_per_blk;
        proj_wmma_kernel<<<grid, 256, 0, stream>>>(dist, W, bias, out, ntiles);
    }
}